// FinalMPNNModel_45896020525894
// MI455X (gfx1250) — compile-verified
//
#include <hip/hip_runtime.h>
#include <hip/hip_bf16.h>

// ---------------------------------------------------------------------------
// MPNN (E(n)-GNN) forward for MI455X / gfx1250, f32 WMMA (V_WMMA_F32_16X16X4_F32)
//
// Roofline: ~70 GF f32 matmul vs ~2.5-3 GB intermediate traffic (t1/t2 must be
// materialized because batch-norm needs full-batch stats before the relu that
// feeds the next GEMM) + 400 MB output. Both ~100us-class -> keep f32 (matches
// reference numerics), use WMMA f32 16x16x4 for every GEMM.
// ---------------------------------------------------------------------------

#define NN_NODES 10000
#define NE_EDGES 320000
#define STATS_BLOCKS 512
#define CDIV(a, b) (((a) + (b) - 1) / (b))

typedef float v2f __attribute__((ext_vector_type(2)));
typedef float v8f __attribute__((ext_vector_type(8)));

// ---------------------------------------------------------------------------
// Wave-level WMMA f32 GEMM over a 16-row LDS-staged A tile.
//  A: 16 x kpad (row stride kstride, kstride odd -> conflict-free frag loads)
//  B: kpad x ncols row-major in global (ldb)
//  D[16 x ncols] -> out rows [rowBase, rowBase+16)
// A-frag layout (32b A 16x4): lane m=l&15 holds M=m; VGPR v holds K=k0+v+2*(l>=16).
// B-frag layout (4x16): VGPR v lanes hold row K=k0+v+2*(l>=16), col N=nt+(l&15).
// C/D layout: VGPR v = row v + 8*(l>=16), col = nt+(l&15).
// ---------------------------------------------------------------------------
__device__ __forceinline__ void wmma_rows16(const float* A, int kpad, int kstride,
                                            const float* __restrict__ B, int ldb,
                                            int ncols, const float* __restrict__ bias,
                                            float* __restrict__ out, int ldo,
                                            int rowBase, int doRelu) {
  const int lane = threadIdx.x & 31;
  const int hi = lane >> 4;
  const int m = lane & 15;
  for (int nt = 0; nt < ncols; nt += 16) {
    v8f acc;
    const float c0 = bias ? bias[nt + m] : 0.0f;  // bias is per-column N
#pragma unroll
    for (int v = 0; v < 8; ++v) acc[v] = c0;
    for (int k0 = 0; k0 < kpad; k0 += 4) {
      const int ka = k0 + 2 * hi;
      v2f av, bv;
      av.x = A[m * kstride + ka];
      av.y = A[m * kstride + ka + 1];
      bv.x = B[(size_t)ka * ldb + nt + m];
      bv.y = B[(size_t)(ka + 1) * ldb + nt + m];
      acc = __builtin_amdgcn_wmma_f32_16x16x4_f32(false, av, false, bv, (short)0, acc,
                                                  false, false);
    }
#pragma unroll
    for (int v = 0; v < 8; ++v) {
      float val = acc[v];
      if (doRelu) val = fmaxf(val, 0.0f);
      out[(size_t)(rowBase + v + 8 * hi) * ldo + nt + m] = val;
    }
  }
}

__device__ __forceinline__ void lds_fence() {
  // Same-wave DS ops are in-order (ISA 7.3) but make it explicit + pin compiler.
  asm volatile("s_wait_dscnt 0" ::: "memory");
}

// ---------------------------------------------------------------------------
// Elementwise / reduction kernels
// ---------------------------------------------------------------------------
__global__ void mpnn_zero(float* __restrict__ p, size_t n) {
  size_t i = (size_t)blockIdx.x * blockDim.x + threadIdx.x;
  const size_t st = (size_t)gridDim.x * blockDim.x;
  for (; i < n; i += st) p[i] = 0.0f;
}

// Deterministic two-stage column stats.  C must be 64 or 128 (pow2, divides 256).
__global__ __launch_bounds__(256) void mpnn_stats_partial(
    const float* __restrict__ t, int M, int C, int ldt,
    float* __restrict__ psum, float* __restrict__ psq) {
  const int tid = threadIdx.x;
  const int c = tid & (C - 1);
  const int g = tid / C;
  const int G = 256 / C;
  const int rowsPer = CDIV(M, (int)gridDim.x);
  const int r0 = blockIdx.x * rowsPer;
  const int r1 = min(M, r0 + rowsPer);
  float s = 0.0f, q = 0.0f;
  for (int r = r0 + g; r < r1; r += G) {
    const float v = t[(size_t)r * ldt + c];
    s += v;
    q += v * v;
  }
  __shared__ float sb[256], qb[256];
  sb[tid] = s;
  qb[tid] = q;
  __syncthreads();
  if (g == 0) {
    for (int j = 1; j < G; ++j) {
      s += sb[j * C + c];
      q += qb[j * C + c];
    }
    psum[blockIdx.x * C + c] = s;
    psq[blockIdx.x * C + c] = q;
  }
}

__global__ void mpnn_stats_finalize(const float* __restrict__ psum,
                                    const float* __restrict__ psq, int nblk, float invM,
                                    const float* __restrict__ glo,
                                    const float* __restrict__ belo,
                                    const float* __restrict__ ghi,
                                    const float* __restrict__ behi, int split,
                                    float* __restrict__ scale, float* __restrict__ bias,
                                    int C) {
  const int c = threadIdx.x;
  if (c >= C) return;
  float s = 0.0f, q = 0.0f;
  for (int b = 0; b < nblk; ++b) {
    s += psum[b * C + c];
    q += psq[b * C + c];
  }
  const float mean = s * invM;
  const float var = fmaxf(q * invM - mean * mean, 0.0f);
  const float inv = rsqrtf(var + 1e-5f);
  const float gg = (c < split) ? glo[c] : ghi[c - split];
  const float be = (c < split) ? belo[c] : behi[c - split];
  scale[c] = gg * inv;
  bias[c] = be - mean * gg * inv;
}

__global__ void mpnn_bn_apply(const float* __restrict__ in, const float* __restrict__ scale,
                              const float* __restrict__ bias, float* __restrict__ out, int n) {
  const int i = blockIdx.x * blockDim.x + threadIdx.x;
  if (i >= n) return;
  const int c = i & 63;
  out[i] = in[i] * scale[c] + bias[c];
}

__global__ void mpnn_pos_init(const float* __restrict__ x, float* __restrict__ pos) {
  const int i = blockIdx.x * blockDim.x + threadIdx.x;
  if (i >= NN_NODES * 2) return;
  pos[i] = x[(i >> 1) * 6 + (i & 1)];
}

__global__ void mpnn_cnt_cent(const int* __restrict__ src, const int* __restrict__ dst,
                              const float* __restrict__ pos, float* __restrict__ cnt,
                              float* __restrict__ cent) {
  const int e = blockIdx.x * blockDim.x + threadIdx.x;
  if (e >= NE_EDGES) return;
  const int s = src[e], d = dst[e];
  atomicAdd(&cnt[d], 1.0f);
  atomicAdd(&cent[d * 2 + 0], pos[s * 2 + 0]);
  atomicAdd(&cent[d * 2 + 1], pos[s * 2 + 1]);
}

__global__ void mpnn_cent_div(const float* __restrict__ cnt, float* __restrict__ cent) {
  const int i = blockIdx.x * blockDim.x + threadIdx.x;
  if (i >= NN_NODES) return;
  const float c = fmaxf(cnt[i], 1.0f);
  cent[i * 2 + 0] /= c;
  cent[i * 2 + 1] /= c;
}

// segment_max (relu'd values >= 0 -> int-bit atomicMax valid, init 0 covers
// the isneginf->0 rule) + msg_p segment_sum.
__global__ void mpnn_agg(const float* __restrict__ t2, const float* __restrict__ scaleB,
                         const float* __restrict__ biasB, const float* __restrict__ posb2,
                         const float* __restrict__ pos, const int* __restrict__ src,
                         const int* __restrict__ dst, float* __restrict__ agg1,
                         float* __restrict__ agg2) {
  const long gid = (long)blockIdx.x * blockDim.x + threadIdx.x;
  if (gid >= (long)NE_EDGES * 64) return;
  const int e = (int)(gid >> 6);
  const int c = (int)(gid & 63);
  const int d = dst[e];
  float v = t2[(size_t)e * 80 + c] * scaleB[c] + biasB[c];
  v = fmaxf(v, 0.0f);
  atomicMax((int*)(agg1 + (size_t)d * 64 + c), __float_as_int(v));
  if (c < 2) {
    const int s = src[e];
    const float stot = t2[(size_t)e * 80 + 64] + posb2[0];
    atomicAdd(&agg2[d * 2 + c], stot * (pos[s * 2 + c] - pos[d * 2 + c]));
  }
}

__global__ void mpnn_layer_fin(const float* __restrict__ hbn, const float* __restrict__ u2,
                               const float* __restrict__ scale, const float* __restrict__ bias,
                               float* __restrict__ h) {
  const int i = blockIdx.x * blockDim.x + threadIdx.x;
  if (i >= NN_NODES * 64) return;
  const int c = i & 63;
  h[i] = hbn[i] + fmaxf(u2[i] * scale[c] + bias[c], 0.0f);
}

__global__ void mpnn_pos_upd(const float* __restrict__ agg2, float* __restrict__ pos) {
  const int i = blockIdx.x * blockDim.x + threadIdx.x;
  if (i >= NN_NODES * 2) return;
  pos[i] += agg2[i];
}

// Packed weights: W1cat 132x128 = [msgW1 | posW1; zero pad rows 130..131]
__global__ void mpnn_build_w1cat(const float* __restrict__ msgW1,
                                 const float* __restrict__ posW1, float* __restrict__ w) {
  const int i = blockIdx.x * blockDim.x + threadIdx.x;
  if (i >= 132 * 128) return;
  const int r = i >> 7, c = i & 127;
  float v = 0.0f;
  if (r < 130) v = (c < 64) ? msgW1[r * 64 + c] : posW1[r * 64 + (c - 64)];
  w[i] = v;
}

// W2cat 128x80 block-diagonal: rows 0..63 cols 0..63 = msgW2; rows 64..127 col 64 = posW2.
__global__ void mpnn_build_w2cat(const float* __restrict__ msgW2,
                                 const float* __restrict__ posW2, float* __restrict__ w) {
  const int i = blockIdx.x * blockDim.x + threadIdx.x;
  if (i >= 128 * 80) return;
  const int r = i / 80, c = i - r * 80;
  float v = 0.0f;
  if (r < 64) {
    if (c < 64) v = msgW2[r * 64 + c];
  } else if (c == 64) {
    v = posW2[r - 64];
  }
  w[i] = v;
}

// ---------------------------------------------------------------------------
// WMMA GEMM kernels (4 waves/block, each wave owns one 16-row tile)
// ---------------------------------------------------------------------------
__global__ __launch_bounds__(128) void mpnn_gemm_init(const float* __restrict__ x,
                                                      const float* __restrict__ linW,
                                                      const float* __restrict__ linb,
                                                      float* __restrict__ h, int ntiles) {
  __shared__ float sA[4][16 * 5];
  const int wave = threadIdx.x >> 5, lane = threadIdx.x & 31;
  const int tile = blockIdx.x * 4 + wave;
  if (tile >= ntiles) return;  // wave-uniform; no cross-wave sync used
  float* A = sA[wave];
  for (int idx = lane; idx < 16 * 4; idx += 32) {
    const int r = idx >> 2, k = idx & 3;
    A[r * 5 + k] = x[(size_t)(tile * 16 + r) * 6 + 2 + k];
  }
  lds_fence();
  wmma_rows16(A, 4, 5, linW, 64, 64, linb, h, 64, tile * 16, 0);
}

// feat = [hbn[dst](64) | hbn[src](64) | d1 | d2 | 0 | 0]  (Kpad=132)
__global__ __launch_bounds__(128) void mpnn_gemm1(
    const float* __restrict__ hbn, const float* __restrict__ pos,
    const float* __restrict__ cent, const int* __restrict__ src,
    const int* __restrict__ dst, const float* __restrict__ w1cat, float* __restrict__ t1) {
  __shared__ float sA[4][16 * 133];
  const int wave = threadIdx.x >> 5, lane = threadIdx.x & 31;
  const int tile = blockIdx.x * 4 + wave;
  float* A = sA[wave];
  for (int idx = lane; idx < 16 * 132; idx += 32) {
    const int r = idx / 132, k = idx - r * 132;
    const int e = tile * 16 + r;
    const int s = src[e], d = dst[e];
    float v;
    if (k < 64) {
      v = hbn[(size_t)d * 64 + k];
    } else if (k < 128) {
      v = hbn[(size_t)s * 64 + (k - 64)];
    } else if (k == 128) {
      const float dx = pos[d * 2] - pos[s * 2], dy = pos[d * 2 + 1] - pos[s * 2 + 1];
      v = sqrtf(dx * dx + dy * dy);
    } else if (k == 129) {
      const float dx = pos[s * 2] - cent[d * 2], dy = pos[s * 2 + 1] - cent[d * 2 + 1];
      v = sqrtf(dx * dx + dy * dy);
    } else {
      v = 0.0f;
    }
    A[r * 133 + k] = v;
  }
  lds_fence();
  wmma_rows16(A, 132, 133, w1cat, 128, 128, nullptr, t1, 128, tile * 16, 0);
}

// A = relu(bn(t1)) fused at load; B = block-diagonal W2cat (128x80)
__global__ __launch_bounds__(128) void mpnn_gemm2(const float* __restrict__ t1,
                                                  const float* __restrict__ scale,
                                                  const float* __restrict__ bias,
                                                  const float* __restrict__ w2cat,
                                                  float* __restrict__ t2) {
  __shared__ float sA[4][16 * 129];
  const int wave = threadIdx.x >> 5, lane = threadIdx.x & 31;
  const int tile = blockIdx.x * 4 + wave;
  float* A = sA[wave];
  for (int idx = lane; idx < 16 * 128; idx += 32) {
    const int r = idx >> 7, k = idx & 127;
    const float v = t1[(size_t)(tile * 16 + r) * 128 + k] * scale[k] + bias[k];
    A[r * 129 + k] = fmaxf(v, 0.0f);
  }
  lds_fence();
  wmma_rows16(A, 128, 129, w2cat, 80, 80, nullptr, t2, 80, tile * 16, 0);
}

// A = [hbn | agg1] per node (K=128), B = upd.W1 (128x64, direct)
__global__ __launch_bounds__(128) void mpnn_gemm_upd1(const float* __restrict__ hbn,
                                                      const float* __restrict__ agg1,
                                                      const float* __restrict__ W,
                                                      float* __restrict__ u1, int ntiles) {
  __shared__ float sA[4][16 * 129];
  const int wave = threadIdx.x >> 5, lane = threadIdx.x & 31;
  const int tile = blockIdx.x * 4 + wave;
  if (tile >= ntiles) return;
  float* A = sA[wave];
  for (int idx = lane; idx < 16 * 128; idx += 32) {
    const int r = idx >> 7, k = idx & 127;
    const int node = tile * 16 + r;
    A[r * 129 + k] = (k < 64) ? hbn[(size_t)node * 64 + k] : agg1[(size_t)node * 64 + (k - 64)];
  }
  lds_fence();
  wmma_rows16(A, 128, 129, W, 64, 64, nullptr, u1, 64, tile * 16, 0);
}

// A = relu(bn(u1)) (K=64), B = upd.W2 (64x64, direct)
__global__ __launch_bounds__(128) void mpnn_gemm_upd2(const float* __restrict__ u1,
                                                      const float* __restrict__ scale,
                                                      const float* __restrict__ bias,
                                                      const float* __restrict__ W,
                                                      float* __restrict__ u2, int ntiles) {
  __shared__ float sA[4][16 * 65];
  const int wave = threadIdx.x >> 5, lane = threadIdx.x & 31;
  const int tile = blockIdx.x * 4 + wave;
  if (tile >= ntiles) return;
  float* A = sA[wave];
  for (int idx = lane; idx < 16 * 64; idx += 32) {
    const int r = idx >> 6, k = idx & 63;
    const float v = u1[(size_t)(tile * 16 + r) * 64 + k] * scale[k] + bias[k];
    A[r * 65 + k] = fmaxf(v, 0.0f);
  }
  lds_fence();
  wmma_rows16(A, 64, 65, W, 64, 64, nullptr, u2, 64, tile * 16, 0);
}

// hcat = [h[src] | h[dst]] (K=128), B = mW1 (128x32), bias mb1, relu -> t3
__global__ __launch_bounds__(128) void mpnn_gemm_final(const float* __restrict__ h,
                                                       const int* __restrict__ src,
                                                       const int* __restrict__ dst,
                                                       const float* __restrict__ mW1,
                                                       const float* __restrict__ mb1,
                                                       float* __restrict__ t3) {
  __shared__ float sA[4][16 * 129];
  const int wave = threadIdx.x >> 5, lane = threadIdx.x & 31;
  const int tile = blockIdx.x * 4 + wave;
  float* A = sA[wave];
  for (int idx = lane; idx < 16 * 128; idx += 32) {
    const int r = idx >> 7, k = idx & 127;
    const int e = tile * 16 + r;
    A[r * 129 + k] =
        (k < 64) ? h[(size_t)src[e] * 64 + k] : h[(size_t)dst[e] * 64 + (k - 64)];
  }
  lds_fence();
  wmma_rows16(A, 128, 129, mW1, 32, 32, mb1, t3, 32, tile * 16, 1);
}

// e = t3[e,:] . mW2 + mb2 ; E[src,dst] += e
__global__ void mpnn_edge_out(const float* __restrict__ t3, const float* __restrict__ mW2,
                              const float* __restrict__ mb2, const int* __restrict__ src,
                              const int* __restrict__ dst, float* __restrict__ out) {
  const int e = blockIdx.x * blockDim.x + threadIdx.x;
  if (e >= NE_EDGES) return;
  float acc = mb2[0];
#pragma unroll
  for (int j = 0; j < 32; ++j) acc += t3[(size_t)e * 32 + j] * mW2[j];
  atomicAdd(&out[(size_t)src[e] * NN_NODES + dst[e]], acc);
}

// ---------------------------------------------------------------------------
// Host side
// ---------------------------------------------------------------------------
struct LayerP {
  const float *msgW1, *msgg1, *msgbe1, *msgW2, *msgg2, *msgbe2;
  const float *updW1, *updg1, *updbe1, *updW2, *updg2, *updbe2;
  const float *posW1, *posg1, *posbe1, *posW2, *posb2;
};

extern "C" void kernel_launch(void* const* d_in, const int* in_sizes, int n_in,
                              void* d_out, int out_size, void* d_ws, size_t ws_size,
                              hipStream_t stream) {
  const int NN = NN_NODES, NE = NE_EDGES;
  (void)ws_size;

  // --- input mapping: support insertion-order and jax-sorted-leaf flattening ---
  int xIdx = 0, eIdx = n_in - 1;
  if (in_sizes[0] == 2 * NE) { eIdx = 0; xIdx = n_in - 1; }  // sorted top-level
  const float* x = (const float*)d_in[xIdx];
  const int* ei = (const int*)d_in[eIdx];
  const int *src = ei, *dst = ei + NE;
  const int p0 = 1;  // params always occupy the middle 118 slots
  auto F = [&](int i) { return (const float*)d_in[i]; };
  const bool sorted = (in_sizes[p0] != 4 * 64);  // insertion order starts with lin_W (256)

  const float *linW, *linb, *bng, *bnb, *mW1, *mb1, *mW2, *mb2;
  LayerP lp[5];
  if (!sorted) {
    linW = F(p0 + 0); linb = F(p0 + 1); bng = F(p0 + 2); bnb = F(p0 + 3);
    for (int L = 0; L < 5; ++L) {
      const int b = p0 + 4 + L * 22;
      lp[L].msgW1 = F(b + 0); lp[L].msgg1 = F(b + 2); lp[L].msgbe1 = F(b + 3);
      lp[L].msgW2 = F(b + 4); lp[L].msgg2 = F(b + 6); lp[L].msgbe2 = F(b + 7);
      lp[L].updW1 = F(b + 8); lp[L].updg1 = F(b + 10); lp[L].updbe1 = F(b + 11);
      lp[L].updW2 = F(b + 12); lp[L].updg2 = F(b + 14); lp[L].updbe2 = F(b + 15);
      lp[L].posW1 = F(b + 16); lp[L].posg1 = F(b + 18); lp[L].posbe1 = F(b + 19);
      lp[L].posW2 = F(b + 20); lp[L].posb2 = F(b + 21);
    }
    const int t = p0 + 4 + 110;
    mW1 = F(t + 0); mb1 = F(t + 1); mW2 = F(t + 2); mb2 = F(t + 3);
  } else {
    // sorted keys: bn_b, bn_g, layers[ msg{W1,W2,b1,b2,be1,be2,g1,g2},
    //              pos{W1,W2,b1,b2,be1,g1}, upd{...} ], lin_W, lin_b, mW1, mW2, mb1, mb2
    bnb = F(p0 + 0); bng = F(p0 + 1);
    for (int L = 0; L < 5; ++L) {
      const int b = p0 + 2 + L * 22;
      lp[L].msgW1 = F(b + 0); lp[L].msgW2 = F(b + 1); lp[L].msgbe1 = F(b + 4);
      lp[L].msgbe2 = F(b + 5); lp[L].msgg1 = F(b + 6); lp[L].msgg2 = F(b + 7);
      lp[L].posW1 = F(b + 8); lp[L].posW2 = F(b + 9); lp[L].posb2 = F(b + 11);
      lp[L].posbe1 = F(b + 12); lp[L].posg1 = F(b + 13);
      lp[L].updW1 = F(b + 14); lp[L].updW2 = F(b + 15); lp[L].updbe1 = F(b + 18);
      lp[L].updbe2 = F(b + 19); lp[L].updg1 = F(b + 20); lp[L].updg2 = F(b + 21);
    }
    const int t = p0 + 2 + 110;
    linW = F(t + 0); linb = F(t + 1); mW1 = F(t + 2); mW2 = F(t + 3);
    mb1 = F(t + 4); mb2 = F(t + 5);
  }

  // --- workspace bump allocator (floats, 256B aligned) ---
  float* ws = (float*)d_ws;
  size_t off = 0;
  auto WA = [&](size_t n) {
    float* p = ws + off;
    off += (n + 63) & ~(size_t)63;
    return p;
  };
  float* h = WA((size_t)NN * 64);
  float* hbn = WA((size_t)NN * 64);
  float* pos = WA((size_t)NN * 2);
  float* cnt = WA(NN);
  float* cent = WA((size_t)NN * 2);
  float* agg1 = WA((size_t)NN * 64);
  float* agg2 = WA((size_t)NN * 2);
  float* u1 = WA((size_t)NN * 64);
  float* u2 = WA((size_t)NN * 64);
  float* scaleH = WA(64); float* biasH = WA(64);
  float* scaleA = WA(128); float* biasA = WA(128);
  float* scaleB = WA(64); float* biasB = WA(64);
  float* scaleU1 = WA(64); float* biasU1 = WA(64);
  float* scaleU2 = WA(64); float* biasU2 = WA(64);
  float* psum = WA((size_t)STATS_BLOCKS * 128);
  float* psq = WA((size_t)STATS_BLOCKS * 128);
  float* w1cat = WA((size_t)5 * 132 * 128);
  float* w2cat = WA((size_t)5 * 128 * 80);
  float* t1 = WA((size_t)NE * 128);
  float* t2 = WA((size_t)NE * 80);
  float* t3 = t1;  // t1 dead after layer loop; reuse for final-MLP hidden

  const int ntilesN = NN / 16;            // 625
  const int egBlocks = NE / 16 / 4;       // 5000 blocks, 4 waves each
  const int ndBlocks = CDIV(ntilesN, 4);  // 157

  // zero dense output (poisoned by harness)
  mpnn_zero<<<4096, 256, 0, stream>>>((float*)d_out, (size_t)out_size);

  // pack per-layer weights (recomputed each call; no cross-call state)
  for (int L = 0; L < 5; ++L) {
    mpnn_build_w1cat<<<CDIV(132 * 128, 256), 256, 0, stream>>>(lp[L].msgW1, lp[L].posW1,
                                                               w1cat + (size_t)L * 132 * 128);
    mpnn_build_w2cat<<<CDIV(128 * 80, 256), 256, 0, stream>>>(lp[L].msgW2, lp[L].posW2,
                                                              w2cat + (size_t)L * 128 * 80);
  }

  // h0 = x[:,2:6] @ lin_W + lin_b (WMMA, K=4 exactly one frag); pos = x[:,:2]
  mpnn_gemm_init<<<ndBlocks, 128, 0, stream>>>(x, linW, linb, h, ntilesN);
  mpnn_pos_init<<<CDIV(NN * 2, 256), 256, 0, stream>>>(x, pos);

  for (int L = 0; L < 5; ++L) {
    const LayerP& P = lp[L];
    // hbn = BN(h; bn_g, bn_b)
    mpnn_stats_partial<<<STATS_BLOCKS, 256, 0, stream>>>(h, NN, 64, 64, psum, psq);
    mpnn_stats_finalize<<<1, 64, 0, stream>>>(psum, psq, STATS_BLOCKS, 1.0f / NN, bng, bnb,
                                              nullptr, nullptr, 64, scaleH, biasH, 64);
    mpnn_bn_apply<<<CDIV(NN * 64, 256), 256, 0, stream>>>(h, scaleH, biasH, hbn, NN * 64);

    // centroids + clear aggregation buffers
    mpnn_zero<<<64, 256, 0, stream>>>(cnt, NN);
    mpnn_zero<<<64, 256, 0, stream>>>(cent, (size_t)NN * 2);
    mpnn_zero<<<2500, 256, 0, stream>>>(agg1, (size_t)NN * 64);
    mpnn_zero<<<64, 256, 0, stream>>>(agg2, (size_t)NN * 2);
    mpnn_cnt_cent<<<CDIV(NE, 256), 256, 0, stream>>>(src, dst, pos, cnt, cent);
    mpnn_cent_div<<<CDIV(NN, 256), 256, 0, stream>>>(cnt, cent);

    // t1 = feat @ [msgW1 | posW1]   (biases cancel through BN)
    mpnn_gemm1<<<egBlocks, 128, 0, stream>>>(hbn, pos, cent, src, dst,
                                             w1cat + (size_t)L * 132 * 128, t1);
    mpnn_stats_partial<<<STATS_BLOCKS, 256, 0, stream>>>(t1, NE, 128, 128, psum, psq);
    mpnn_stats_finalize<<<1, 128, 0, stream>>>(psum, psq, STATS_BLOCKS, 1.0f / NE, P.msgg1,
                                               P.msgbe1, P.posg1, P.posbe1, 64, scaleA,
                                               biasA, 128);

    // t2 = relu(bn(t1)) @ blockdiag(msgW2, posW2)   (col 64 = pos scalar pre-bias)
    mpnn_gemm2<<<egBlocks, 128, 0, stream>>>(t1, scaleA, biasA,
                                             w2cat + (size_t)L * 128 * 80, t2);
    mpnn_stats_partial<<<STATS_BLOCKS, 256, 0, stream>>>(t2, NE, 64, 80, psum, psq);
    mpnn_stats_finalize<<<1, 64, 0, stream>>>(psum, psq, STATS_BLOCKS, 1.0f / NE, P.msgg2,
                                              P.msgbe2, nullptr, nullptr, 64, scaleB, biasB,
                                              64);

    // agg1 = segmax(relu(bn(t2))), agg2 = segsum(s * (pos_j - pos_i))
    mpnn_agg<<<CDIV(NE * 64, 256), 256, 0, stream>>>(t2, scaleB, biasB, P.posb2, pos, src,
                                                     dst, agg1, agg2);

    // update block: u1 = [hbn|agg1] @ updW1 ; u2 = relu(bn(u1)) @ updW2
    mpnn_gemm_upd1<<<ndBlocks, 128, 0, stream>>>(hbn, agg1, P.updW1, u1, ntilesN);
    mpnn_stats_partial<<<STATS_BLOCKS, 256, 0, stream>>>(u1, NN, 64, 64, psum, psq);
    mpnn_stats_finalize<<<1, 64, 0, stream>>>(psum, psq, STATS_BLOCKS, 1.0f / NN, P.updg1,
                                              P.updbe1, nullptr, nullptr, 64, scaleU1,
                                              biasU1, 64);
    mpnn_gemm_upd2<<<ndBlocks, 128, 0, stream>>>(u1, scaleU1, biasU1, P.updW2, u2, ntilesN);
    mpnn_stats_partial<<<STATS_BLOCKS, 256, 0, stream>>>(u2, NN, 64, 64, psum, psq);
    mpnn_stats_finalize<<<1, 64, 0, stream>>>(psum, psq, STATS_BLOCKS, 1.0f / NN, P.updg2,
                                              P.updbe2, nullptr, nullptr, 64, scaleU2,
                                              biasU2, 64);

    // h = hbn + relu(bn(u2)) ; pos += agg2
    mpnn_layer_fin<<<CDIV(NN * 64, 256), 256, 0, stream>>>(hbn, u2, scaleU2, biasU2, h);
    mpnn_pos_upd<<<CDIV(NN * 2, 256), 256, 0, stream>>>(agg2, pos);
  }

  // final edge MLP: t3 = relu([h_src|h_dst] @ mW1 + mb1); E[src,dst] += t3 . mW2 + mb2
  mpnn_gemm_final<<<egBlocks, 128, 0, stream>>>(h, src, dst, mW1, mb1, t3);
  mpnn_edge_out<<<CDIV(NE, 256), 256, 0, stream>>>(t3, mW2, mb2, src, dst, (float*)d_out);
}